// myAttentionCostAggregation_65335042506912
// MI455X (gfx1250) — compile-verified
//
#include <hip/hip_runtime.h>
#include <cstddef>

typedef __attribute__((ext_vector_type(2))) float v2f;
typedef __attribute__((ext_vector_type(8))) float v8f;

static constexpr int Hh = 48;
static constexpr int Ww = 96;
static constexpr int Dd = 32;
static constexpr int Cc = 64;
static constexpr int HW = Hh * Ww;   // 4608

static __device__ __forceinline__ v8f wmma_f32(v2f a, v2f b, v8f c) {
  // D = A(16x4) x B(4x16) + C, fp32 exact
  return __builtin_amdgcn_wmma_f32_16x16x4_f32(false, a, false, b, (short)0, c,
                                               false, false);
}

// ---------------------------------------------------------------------------
// conv3x3 (pad=1, cross-correlation, 32->32 ch) + batchnorm
// grid: (Dd*Hh) blocks, Ww threads
// ---------------------------------------------------------------------------
__global__ void conv3x3_bn_kernel(const float* __restrict__ x,     // [32][48][96]
                                  const float* __restrict__ w,     // [32][32][3][3]
                                  const float* __restrict__ scale,
                                  const float* __restrict__ bias,
                                  const float* __restrict__ mean,
                                  const float* __restrict__ var,
                                  float* __restrict__ y) {
  const int wcol = threadIdx.x;            // 0..95
  const int blk  = blockIdx.x;             // d*H + i
  const int d = blk / Hh, i = blk - d * Hh;
  float acc = 0.f;
  for (int c = 0; c < Dd; ++c) {
    const float* xc = x + (size_t)c * HW;
    const float* wc = w + (size_t)(d * Dd + c) * 9;
#pragma unroll
    for (int dy = -1; dy <= 1; ++dy) {
      int yy = i + dy;
      if (yy < 0 || yy >= Hh) continue;
#pragma unroll
      for (int dx = -1; dx <= 1; ++dx) {
        int xx = wcol + dx;
        if (xx < 0 || xx >= Ww) continue;
        acc += xc[yy * Ww + xx] * wc[(dy + 1) * 3 + (dx + 1)];
      }
    }
  }
  float inv = scale[d] / sqrtf(var[d] + 1e-5f);
  y[(size_t)(d * Hh + i) * Ww + wcol] = (acc - mean[d]) * inv + bias[d];
}

// ---------------------------------------------------------------------------
// 1x1 conv as GEMM via WMMA f32 16x16x4:
//   Y[o][p] = sum_c Wm[o][c] * X[c][p] + Bv[o],  p = 0..4607
// One wave computes a 16(M=o) x 16(N=p) tile. 128 thr = 4 waves / block.
// grid = (Cout/16)*(4608/16)/4 blocks (exact).
// ---------------------------------------------------------------------------
__global__ void conv1x1_wmma_kernel(const float* __restrict__ X,
                                    const float* __restrict__ Wm,
                                    const float* __restrict__ Bv,
                                    float* __restrict__ Y,
                                    int Cout, int Cin) {
  const int N = HW;
  const int NT = N / 16;                    // 288
  const int lane = threadIdx.x & 31;
  const int wid  = threadIdx.x >> 5;
  const int tile = blockIdx.x * 4 + wid;
  const int mt = tile / NT, nt = tile - mt * NT;
  const int m0 = mt * 16, p0 = nt * 16;
  const int hi = lane >> 4, l = lane & 15;
  v8f c = {};
  for (int k0 = 0; k0 < Cin; k0 += 4) {
    int ka = k0 + hi * 2;                   // A/B K layout: lanes0-15 {k0,k0+1}, lanes16-31 {k0+2,k0+3}
    v2f a, b;
    a.x = Wm[(size_t)(m0 + l) * Cin + ka];
    a.y = Wm[(size_t)(m0 + l) * Cin + ka + 1];
    b.x = X[(size_t)(ka)     * N + p0 + l];
    b.y = X[(size_t)(ka + 1) * N + p0 + l];
    c = wmma_f32(a, b, c);
  }
#pragma unroll
  for (int r = 0; r < 8; ++r) {
    int row = m0 + r + hi * 8;              // C/D layout: VGPR r -> M = r (+8 for high lanes)
    Y[(size_t)row * N + p0 + l] = c[r] + Bv[row];
  }
}

// ---------------------------------------------------------------------------
// eW logits: per block one (d, i).  E[w][v] = sum_c q[c,i,w] * mixed[d,c,i,v]
// mixed[d,c,i,v] = lk[c,i,v] + (v-d>=0 ? rk[c,i,v-d] : 0)
// bufW layout: [d][i][w][v]
// ---------------------------------------------------------------------------
__global__ void ew_kernel(const float* __restrict__ q, const float* __restrict__ lk,
                          const float* __restrict__ rk, float* __restrict__ bufW) {
  __shared__ float Qs[Cc * Ww];             // 24 KB
  __shared__ float Ms[Cc * Ww];             // 24 KB
  const int i = blockIdx.x, d = blockIdx.y;
  for (int t = threadIdx.x; t < Cc * Ww; t += blockDim.x) {
    int c = t / Ww, v = t - c * Ww;
    size_t off = (size_t)c * HW + i * Ww + v;
    Qs[t] = q[off];
    float m = lk[off];
    int s = v - d;
    if (s >= 0) m += rk[(size_t)c * HW + i * Ww + s];
    Ms[t] = m;
  }
  __syncthreads();
  const int lane = threadIdx.x & 31, wid = threadIdx.x >> 5;
  const int hi = lane >> 4, l = lane & 15;
  float* dst = bufW + (size_t)(d * Hh + i) * Ww * Ww;
  for (int tile = wid; tile < 36; tile += 4) {
    int wt = tile / 6, vt = tile - wt * 6;
    int w0 = wt * 16, v0 = vt * 16;
    v8f c = {};
#pragma unroll
    for (int k0 = 0; k0 < Cc; k0 += 4) {
      int ka = k0 + hi * 2;
      v2f a, b;
      a.x = Qs[(ka)     * Ww + w0 + l];     // A[m=w][k=c] = Q[c][w]
      a.y = Qs[(ka + 1) * Ww + w0 + l];
      b.x = Ms[(ka)     * Ww + v0 + l];     // B[k=c][n=v] = M[c][v]
      b.y = Ms[(ka + 1) * Ww + v0 + l];
      c = wmma_f32(a, b, c);
    }
#pragma unroll
    for (int r = 0; r < 8; ++r)
      dst[(size_t)(w0 + r + hi * 8) * Ww + v0 + l] = c[r];
  }
}

// ---------------------------------------------------------------------------
// eH logits: per block one (d, w).  E[i][j] = sum_c q[c,i,w] * mixed[d,c,j,w]
// mixed[d,c,j,w] = lk[c,j,w] + (w-d>=0 ? rk[c,j,w-d] : 0);  diag(i==j) = -inf
// bufH layout: [d][w][i][j]
// ---------------------------------------------------------------------------
__global__ void eh_kernel(const float* __restrict__ q, const float* __restrict__ lk,
                          const float* __restrict__ rk, float* __restrict__ bufH) {
  __shared__ float Qs[Cc * Hh];             // 12 KB
  __shared__ float Ms[Cc * Hh];             // 12 KB
  const int w = blockIdx.x, d = blockIdx.y;
  const int shifted = w - d;
  for (int t = threadIdx.x; t < Cc * Hh; t += blockDim.x) {
    int c = t / Hh, j = t - c * Hh;
    size_t off = (size_t)c * HW + j * Ww + w;
    Qs[t] = q[off];
    float m = lk[off];
    if (shifted >= 0) m += rk[(size_t)c * HW + j * Ww + shifted];
    Ms[t] = m;
  }
  __syncthreads();
  const int lane = threadIdx.x & 31, wid = threadIdx.x >> 5;
  const int hi = lane >> 4, l = lane & 15;
  float* dst = bufH + (size_t)(d * Ww + w) * Hh * Hh;
  for (int tile = wid; tile < 9; tile += 4) {
    int it = tile / 3, jt = tile - it * 3;
    int i0 = it * 16, j0 = jt * 16;
    v8f c = {};
#pragma unroll
    for (int k0 = 0; k0 < Cc; k0 += 4) {
      int ka = k0 + hi * 2;
      v2f a, b;
      a.x = Qs[(ka)     * Hh + i0 + l];
      a.y = Qs[(ka + 1) * Hh + i0 + l];
      b.x = Ms[(ka)     * Hh + j0 + l];
      b.y = Ms[(ka + 1) * Hh + j0 + l];
      c = wmma_f32(a, b, c);
    }
#pragma unroll
    for (int r = 0; r < 8; ++r) {
      int row = i0 + r + hi * 8;            // i
      int col = j0 + l;                     // j
      float val = (row == col) ? -__builtin_inff() : c[r];
      dst[(size_t)row * Hh + col] = val;
    }
  }
}

// ---------------------------------------------------------------------------
// softmax over concat(eH row[48], eW row[96]) per (d,i,w); in-place -> att
// one wave per row; 147456 rows.
// ---------------------------------------------------------------------------
__global__ void softmax_kernel(float* __restrict__ bufH, float* __restrict__ bufW) {
  const int row  = (blockIdx.x * blockDim.x + threadIdx.x) >> 5;
  const int lane = threadIdx.x & 31;
  const int d = row / (Hh * Ww);
  const int rem = row - d * (Hh * Ww);
  const int i = rem / Ww, w = rem - i * Ww;
  float* ph = bufH + ((size_t)(d * Ww + w) * Hh + i) * Hh;   // 48
  float* pw = bufW + ((size_t)(d * Hh + i) * Ww + w) * Ww;   // 96
  float h0 = ph[lane];
  float h1 = (lane < 16) ? ph[32 + lane] : -__builtin_inff();
  float w0 = pw[lane], w1 = pw[32 + lane], w2 = pw[64 + lane];
  float m = fmaxf(fmaxf(fmaxf(h0, h1), fmaxf(w0, w1)), w2);
#pragma unroll
  for (int s = 16; s; s >>= 1) m = fmaxf(m, __shfl_xor(m, s));
  float eh0 = __expf(h0 - m);                       // exp(-inf)=0 on diagonal
  float eh1 = (lane < 16) ? __expf(h1 - m) : 0.f;
  float ew0 = __expf(w0 - m), ew1 = __expf(w1 - m), ew2 = __expf(w2 - m);
  float s = eh0 + eh1 + ew0 + ew1 + ew2;
#pragma unroll
  for (int sh = 16; sh; sh >>= 1) s += __shfl_xor(s, sh);
  float inv = 1.f / s;
  ph[lane] = eh0 * inv;
  if (lane < 16) ph[32 + lane] = eh1 * inv;
  pw[lane] = ew0 * inv;
  pw[32 + lane] = ew1 * inv;
  pw[64 + lane] = ew2 * inv;
}

// ---------------------------------------------------------------------------
// out[d,i,w] += gamma[d] * ( sum_j v[d,j,w]*attH[d,i,w,j]
//                          + sum_vc v[d,i,vc]*attW[d,i,w,vc] )
// one wave per output element.
// ---------------------------------------------------------------------------
__global__ void aggregate_kernel(const float* __restrict__ vbuf,
                                 const float* __restrict__ attH,
                                 const float* __restrict__ attW,
                                 const float* __restrict__ gammas,
                                 float* __restrict__ out) {
  const int row  = (blockIdx.x * blockDim.x + threadIdx.x) >> 5;
  const int lane = threadIdx.x & 31;
  const int d = row / (Hh * Ww);
  const int rem = row - d * (Hh * Ww);
  const int i = rem / Ww, w = rem - i * Ww;
  const float* ah = attH + ((size_t)(d * Ww + w) * Hh + i) * Hh;
  const float* aw = attW + ((size_t)(d * Hh + i) * Ww + w) * Ww;
  const float* vd = vbuf + (size_t)d * HW;
  float s = ah[lane] * vd[lane * Ww + w];
  if (lane < 16) s += ah[32 + lane] * vd[(32 + lane) * Ww + w];
  const float* vrow = vd + i * Ww;
  s += aw[lane] * vrow[lane] + aw[32 + lane] * vrow[32 + lane]
     + aw[64 + lane] * vrow[64 + lane];
#pragma unroll
  for (int sh = 16; sh; sh >>= 1) s += __shfl_xor(s, sh);
  if (lane == 0) out[(size_t)d * HW + i * Ww + w] += gammas[d] * s;
}

// ---------------------------------------------------------------------------
extern "C" void kernel_launch(void* const* d_in, const int* in_sizes, int n_in,
                              void* d_out, int out_size, void* d_ws, size_t ws_size,
                              hipStream_t stream) {
  const float* cost_volume = (const float*)d_in[0];
  const float* left_q      = (const float*)d_in[1];
  const float* left_k      = (const float*)d_in[2];
  const float* right_k     = (const float*)d_in[3];
  const float* conva_w     = (const float*)d_in[4];
  const float* conva_scale = (const float*)d_in[5];
  const float* conva_bias  = (const float*)d_in[6];
  const float* conva_mean  = (const float*)d_in[7];
  const float* conva_var   = (const float*)d_in[8];
  const float* wq          = (const float*)d_in[9];
  const float* bq          = (const float*)d_in[10];
  const float* wlk         = (const float*)d_in[11];
  const float* blk         = (const float*)d_in[12];
  const float* wrk         = (const float*)d_in[13];
  const float* brk         = (const float*)d_in[14];
  const float* wv          = (const float*)d_in[15];
  const float* bv          = (const float*)d_in[16];
  const float* gammas      = (const float*)d_in[17];
  const float* convb_w     = (const float*)d_in[18];
  const float* convb_scale = (const float*)d_in[19];
  const float* convb_bias  = (const float*)d_in[20];
  const float* convb_mean  = (const float*)d_in[21];
  const float* convb_var   = (const float*)d_in[22];

  // workspace layout (floats): ~90 MB total
  float* ws   = (float*)d_ws;
  float* out0 = ws;                                   // 147456
  float* qb   = out0 + (size_t)Dd * HW;               // 294912
  float* lkb  = qb   + (size_t)Cc * HW;               // 294912
  float* rkb  = lkb  + (size_t)Cc * HW;               // 294912
  float* vbuf = rkb  + (size_t)Cc * HW;               // 147456
  float* bufH = vbuf + (size_t)Dd * HW;               // 32*96*48*48 = 7077888
  float* bufW = bufH + (size_t)Dd * Ww * Hh * Hh;     // 32*48*96*96 = 14155776

  // 1) conv_a + BN
  conv3x3_bn_kernel<<<Dd * Hh, Ww, 0, stream>>>(
      cost_volume, conva_w, conva_scale, conva_bias, conva_mean, conva_var, out0);

  // 2) q / lk / rk 1x1 convs (64x64 GEMM over 4608 pixels), WMMA
  const int g64 = (Cc / 16) * (HW / 16) / 4;   // 288 blocks
  conv1x1_wmma_kernel<<<g64, 128, 0, stream>>>(left_q,  wq,  bq,  qb,  Cc, Cc);
  conv1x1_wmma_kernel<<<g64, 128, 0, stream>>>(left_k,  wlk, blk, lkb, Cc, Cc);
  conv1x1_wmma_kernel<<<g64, 128, 0, stream>>>(right_k, wrk, brk, rkb, Cc, Cc);

  // 3) attention logits via batched WMMA GEMMs
  eh_kernel<<<dim3(Ww, Dd), 128, 0, stream>>>(qb, lkb, rkb, bufH);
  ew_kernel<<<dim3(Hh, Dd), 128, 0, stream>>>(qb, lkb, rkb, bufW);

  // 4) joint softmax over 48+96 logits, in place
  const int rows = Dd * Hh * Ww;               // 147456
  softmax_kernel<<<rows / 8, 256, 0, stream>>>(bufH, bufW);

  // 5) recurrence: v = 1x1 conv(out0); out0 += gamma * (attH.v + attW.v)
  const int g32 = (Dd / 16) * (HW / 16) / 4;   // 144 blocks
  for (int it = 0; it < 2; ++it) {
    conv1x1_wmma_kernel<<<g32, 128, 0, stream>>>(out0, wv, bv, vbuf, Dd, Dd);
    aggregate_kernel<<<rows / 8, 256, 0, stream>>>(vbuf, bufH, bufW, gammas, out0);
  }

  // 6) conv_b + BN -> output
  conv3x3_bn_kernel<<<Dd * Hh, Ww, 0, stream>>>(
      out0, convb_w, convb_scale, convb_bias, convb_mean, convb_var, (float*)d_out);
}